// VectorQuantize_25159918420605
// MI455X (gfx1250) — compile-verified
//
#include <hip/hip_runtime.h>

#define DIM       1024
#define N_EMBED   4096
#define N_TOKENS  16384
#define DECAY     0.8f
#define OMD       0.2f   // 1 - DECAY
#define COMMIT    1.0f
#define EPSV      1e-5f

typedef __attribute__((ext_vector_type(2))) float v2f;
typedef __attribute__((ext_vector_type(8))) float v8f;

// ---------------- utility: zero a float region ----------------
__global__ __launch_bounds__(256) void k_zero(float* __restrict__ p, size_t n) {
  size_t i = (size_t)blockIdx.x * 256 + threadIdx.x;
  size_t stride = (size_t)gridDim.x * 256;
  for (; i < n; i += stride) p[i] = 0.0f;
}

// ---------------- ||e_j||^2 : column sums of embed [DIM, N_EMBED] ----------------
__global__ __launch_bounds__(256) void k_en(const float* __restrict__ e, float* __restrict__ en) {
  int j = blockIdx.x * 256 + threadIdx.x;   // coalesced across j for each d
  float s = 0.0f;
  for (int d = 0; d < DIM; ++d) {
    float v = e[(size_t)d * N_EMBED + j];
    s += v * v;
  }
  en[j] = s;
}

// ---------------- ||x_i||^2 : one wave32 per row ----------------
__global__ __launch_bounds__(256) void k_xn(const float* __restrict__ x, float* __restrict__ xn) {
  int wave = threadIdx.x >> 5;
  int lane = threadIdx.x & 31;
  int row  = blockIdx.x * 8 + wave;
  const float* xr = x + (size_t)row * DIM;
  float s = 0.0f;
  for (int t = lane; t < DIM; t += 32) { float v = xr[t]; s += v * v; }
  #pragma unroll
  for (int off = 16; off; off >>= 1) s += __shfl_xor(s, off, 32);
  if (lane == 0) xn[row] = s;
}

// ---------------- WMMA GEMM: dist = xn + en - 2 * x @ embed ----------------
// 128x128 tile per 256-thread block (8 wave32s). Wave (wm,wn) = (w>>1, w&1)
// owns a 32x64 region = 2(M) x 4(N) subtiles of 16x16.
#define TM 128
#define TN 128
#define KT 32
#define ASTR (KT + 4)    // 36 dwords = 144 B rows (16B aligned, conflict-free)
#define BSTR (TN + 8)    // 136 dwords = 544 B rows; 2*136 mod 64 = 16 -> half-waves hit disjoint banks

__global__ __launch_bounds__(256) void k_gemm_dist(
    const float* __restrict__ x, const float* __restrict__ e,
    const float* __restrict__ xn, const float* __restrict__ en,
    float* __restrict__ dist) {
  __shared__ float Alds[2][TM * ASTR];
  __shared__ float Blds[2][KT * BSTR];

  const int tid  = threadIdx.x;
  const int wave = tid >> 5;
  const int lane = tid & 31;
  const int wm = wave >> 1;   // 0..3 : which 32-row band
  const int wn = wave & 1;    // 0..1 : which 64-col band
  const int m0 = blockIdx.y * TM;
  const int n0 = blockIdx.x * TN;

  v8f acc[2][4] = {};

  const int lrow  = lane & 15;          // M (A) / N (B) within a 16x16 subtile
  const int khalf = (lane >> 4) << 1;   // 0 or 2: K-pair offset for upper half-wave

  auto load_tiles = [&](int buf, int k0) {
    // A tile: 128 x 32 floats = 1024 float4 across 256 threads (4 each)
    #pragma unroll
    for (int t = 0; t < 4; ++t) {
      int idx = tid + t * 256;
      int r   = idx >> 3;      // / (KT/4)
      int c4  = idx & 7;
      *(float4*)&Alds[buf][r * ASTR + (c4 << 2)] =
          *(const float4*)&x[(size_t)(m0 + r) * DIM + k0 + (c4 << 2)];
    }
    // B tile: 32 x 128 floats = 1024 float4
    #pragma unroll
    for (int t = 0; t < 4; ++t) {
      int idx = tid + t * 256;
      int r   = idx >> 5;      // / (TN/4)
      int c4  = idx & 31;
      *(float4*)&Blds[buf][r * BSTR + (c4 << 2)] =
          *(const float4*)&e[(size_t)(k0 + r) * N_EMBED + n0 + (c4 << 2)];
    }
  };

  load_tiles(0, 0);
  int cur = 0;
  for (int k0 = 0; k0 < DIM; k0 += KT) {
    __syncthreads();                       // buf[cur] staged; prior compute of buf[cur] done
    if (k0 + KT < DIM) load_tiles(cur ^ 1, k0 + KT);  // overlap next stage with WMMA

    const float* __restrict__ Ab = &Alds[cur][0];
    const float* __restrict__ Bb = &Blds[cur][0];
    #pragma unroll
    for (int k = 0; k < KT; k += 4) {
      // B 4x16 fragments: lane holds {B[k+khalf][col], B[k+khalf+1][col]}
      v2f b[4];
      #pragma unroll
      for (int nt = 0; nt < 4; ++nt) {
        int col = wn * 64 + nt * 16 + lrow;
        b[nt].x = Bb[(k + khalf)     * BSTR + col];
        b[nt].y = Bb[(k + khalf + 1) * BSTR + col];
      }
      // A 16x4 fragments: lane holds {A[row][k+khalf], A[row][k+khalf+1]}
      v2f a[2];
      #pragma unroll
      for (int mt = 0; mt < 2; ++mt)
        a[mt] = *(const v2f*)&Ab[(wm * 32 + mt * 16 + lrow) * ASTR + k + khalf];
      #pragma unroll
      for (int mt = 0; mt < 2; ++mt)
        #pragma unroll
        for (int nt = 0; nt < 4; ++nt)
          acc[mt][nt] = __builtin_amdgcn_wmma_f32_16x16x4_f32(
              false, a[mt], false, b[nt], (short)0, acc[mt][nt], false, false);
    }
    cur ^= 1;
  }

  // epilogue: C layout — VGPR g: lanes 0-15 -> M=g, lanes 16-31 -> M=g+8
  const int gm0 = m0 + wm * 32 + ((lane >> 4) << 3);
  #pragma unroll
  for (int mt = 0; mt < 2; ++mt) {
    float xnr[8];
    #pragma unroll
    for (int g = 0; g < 8; ++g) xnr[g] = xn[gm0 + mt * 16 + g];
    #pragma unroll
    for (int nt = 0; nt < 4; ++nt) {
      int gcol = n0 + wn * 64 + nt * 16 + lrow;
      float enc = en[gcol];
      #pragma unroll
      for (int g = 0; g < 8; ++g) {
        dist[(size_t)(gm0 + mt * 16 + g) * N_EMBED + gcol] =
            xnr[g] + enc - 2.0f * acc[mt][nt][g];
      }
    }
  }
}

// ---------------- per-token: argmin, gather quantize, loss, counts ----------------
__global__ __launch_bounds__(256) void k_argmin_gather(
    const float* __restrict__ dist, const float* __restrict__ x,
    const float* __restrict__ e,
    float* __restrict__ out_q, float* __restrict__ out_ind,
    int* __restrict__ ind_buf, float* __restrict__ counts,
    float* __restrict__ loss_acc) {
  __shared__ float sbest[256];
  __shared__ int   sidx[256];
  __shared__ int   s_ind;
  const int i   = blockIdx.x;
  const int tid = threadIdx.x;

  const float4* drow4 = (const float4*)(dist + (size_t)i * N_EMBED);
  float best = 3.4e38f;
  int   bidx = 0x7fffffff;
  for (int j4 = tid; j4 < N_EMBED / 4; j4 += 256) {
    float4 v = drow4[j4];
    int base = j4 << 2;
    if (v.x < best) { best = v.x; bidx = base; }
    if (v.y < best) { best = v.y; bidx = base + 1; }
    if (v.z < best) { best = v.z; bidx = base + 2; }
    if (v.w < best) { best = v.w; bidx = base + 3; }
  }
  sbest[tid] = best; sidx[tid] = bidx;
  __syncthreads();
  for (int s = 128; s; s >>= 1) {
    if (tid < s) {
      float ob = sbest[tid + s]; int oi = sidx[tid + s];
      if (ob < sbest[tid] || (ob == sbest[tid] && oi < sidx[tid])) {
        sbest[tid] = ob; sidx[tid] = oi;
      }
    }
    __syncthreads();
  }
  if (tid == 0) {
    int ind = sidx[0];
    s_ind = ind;
    ind_buf[i] = ind;
    out_ind[i] = (float)ind;
    atomicAdd(&counts[ind], 1.0f);
  }
  __syncthreads();
  const int ind = s_ind;

  float lsum = 0.0f;
  for (int d = tid; d < DIM; d += 256) {
    float q  = e[(size_t)d * N_EMBED + ind];   // embed column gather (hot in L2)
    float xv = x[(size_t)i * DIM + d];
    float df = q - xv;
    lsum += df * df;
    out_q[(size_t)i * DIM + d] = xv + (q - xv);  // straight-through estimator
  }
  sbest[tid] = lsum;
  __syncthreads();
  for (int s = 128; s; s >>= 1) {
    if (tid < s) sbest[tid] += sbest[tid + s];
    __syncthreads();
  }
  if (tid == 0) atomicAdd(loss_acc, sbest[0]);
}

// ---------------- segment-sum scatter: embed_sum[d, ind[i]] += x[i, d] ----------------
__global__ __launch_bounds__(256) void k_scatter(
    const float* __restrict__ x, const int* __restrict__ ind_buf,
    float* __restrict__ embed_sum) {
  const int i   = blockIdx.x;
  const int ind = ind_buf[i];
  for (int d = threadIdx.x; d < DIM; d += 256)
    atomicAdd(&embed_sum[(size_t)d * N_EMBED + ind], x[(size_t)i * DIM + d]);
}

// ---------------- EMA cluster size + total + loss scale ----------------
__global__ __launch_bounds__(1024) void k_finish1(
    const float* __restrict__ cluster_size, const float* __restrict__ counts,
    const float* __restrict__ loss_acc,
    float* __restrict__ out_ncs, float* __restrict__ out_loss,
    float* __restrict__ total_ws) {
  __shared__ float sred[1024];
  const int tid = threadIdx.x;
  float s = 0.0f;
  for (int j = tid; j < N_EMBED; j += 1024) {
    float v = DECAY * cluster_size[j] + OMD * counts[j];
    out_ncs[j] = v;
    s += v;
  }
  sred[tid] = s;
  __syncthreads();
  for (int st = 512; st; st >>= 1) {
    if (tid < st) sred[tid] += sred[tid + st];
    __syncthreads();
  }
  if (tid == 0) {
    total_ws[0] = sred[0];
    out_loss[0] = loss_acc[0] / (float)((size_t)N_TOKENS * DIM) * COMMIT;
  }
}

// ---------------- EMA embed_avg + normalized codebook ----------------
__global__ __launch_bounds__(256) void k_finish2(
    const float* __restrict__ embed_avg, const float* __restrict__ embed_sum,
    const float* __restrict__ ncs, const float* __restrict__ total_ws,
    float* __restrict__ out_nea, float* __restrict__ out_enorm) {
  size_t idx = (size_t)blockIdx.x * 256 + threadIdx.x;
  if (idx >= (size_t)DIM * N_EMBED) return;
  int j = (int)(idx & (N_EMBED - 1));
  float total = total_ws[0];
  float na = DECAY * embed_avg[idx] + OMD * embed_sum[idx];
  out_nea[idx] = na;
  float smoothed = (ncs[j] + EPSV) / (total + (float)N_EMBED * EPSV) * total;
  out_enorm[idx] = na / smoothed;
}

extern "C" void kernel_launch(void* const* d_in, const int* in_sizes, int n_in,
                              void* d_out, int out_size, void* d_ws, size_t ws_size,
                              hipStream_t stream) {
  const float* x            = (const float*)d_in[0];  // [16384, 1024]
  const float* embed        = (const float*)d_in[1];  // [1024, 4096]
  const float* cluster_size = (const float*)d_in[2];  // [4096]
  const float* embed_avg    = (const float*)d_in[3];  // [1024, 4096]

  float* out = (float*)d_out;
  // tuple order: quantize_st, embed_ind, loss, dist, new_cluster_size, new_embed_avg, embed_normalized
  float* out_q     = out;                                   // 16384*1024
  float* out_ind   = out_q   + (size_t)N_TOKENS * DIM;      // 16384
  float* out_loss  = out_ind + N_TOKENS;                    // 1
  float* out_dist  = out_loss + 1;                          // 16384*4096
  float* out_ncs   = out_dist + (size_t)N_TOKENS * N_EMBED; // 4096
  float* out_nea   = out_ncs  + N_EMBED;                    // 1024*4096
  float* out_enorm = out_nea  + (size_t)DIM * N_EMBED;      // 1024*4096

  float* ws        = (float*)d_ws;
  float* xn        = ws;                                    // 16384
  float* en        = xn + N_TOKENS;                         // 4096
  float* counts    = en + N_EMBED;                          // 4096
  float* embed_sum = counts + N_EMBED;                      // 1024*4096
  float* loss_acc  = embed_sum + (size_t)DIM * N_EMBED;     // 1
  float* total_ws  = loss_acc + 1;                          // 1
  int*   ind_buf   = (int*)(total_ws + 1);                  // 16384

  // zero the accumulator region (counts | embed_sum | loss_acc | total_ws)
  size_t zero_n = (size_t)N_EMBED + (size_t)DIM * N_EMBED + 2;
  hipLaunchKernelGGL(k_zero, dim3(8192), dim3(256), 0, stream, counts, zero_n);

  hipLaunchKernelGGL(k_en, dim3(N_EMBED / 256), dim3(256), 0, stream, embed, en);
  hipLaunchKernelGGL(k_xn, dim3(N_TOKENS / 8), dim3(256), 0, stream, x, xn);

  hipLaunchKernelGGL(k_gemm_dist, dim3(N_EMBED / TN, N_TOKENS / TM), dim3(256), 0, stream,
                     x, embed, xn, en, out_dist);

  hipLaunchKernelGGL(k_argmin_gather, dim3(N_TOKENS), dim3(256), 0, stream,
                     out_dist, x, embed, out_q, out_ind, ind_buf, counts, loss_acc);

  hipLaunchKernelGGL(k_scatter, dim3(N_TOKENS), dim3(256), 0, stream, x, ind_buf, embed_sum);

  hipLaunchKernelGGL(k_finish1, dim3(1), dim3(1024), 0, stream,
                     cluster_size, counts, loss_acc, out_ncs, out_loss, total_ws);

  size_t n2 = (size_t)DIM * N_EMBED;
  hipLaunchKernelGGL(k_finish2, dim3((unsigned)((n2 + 255) / 256)), dim3(256), 0, stream,
                     embed_avg, embed_sum, out_ncs, total_ws, out_nea, out_enorm);
}